// DecoderObjectCls_14774687498859
// MI455X (gfx1250) — compile-verified
//
#include <hip/hip_runtime.h>
#include <math.h>

typedef float v2f __attribute__((ext_vector_type(2)));
typedef float v8f __attribute__((ext_vector_type(8)));
typedef int   v4i __attribute__((ext_vector_type(4)));

#define B_    32
#define T_    1024
#define N_    100
#define NP_   112        // N padded to 7 tiles of 16
#define HID_  768
#define EMB_  512

#if defined(__has_builtin)
#if __has_builtin(__builtin_amdgcn_global_load_async_to_lds_b128) && \
    __has_builtin(__builtin_amdgcn_s_wait_asynccnt)
#define USE_ASYNC_LDS 1
#endif
#endif

// D = A(16x4 f32) * B(4x16 f32) + C(16x16 f32), full wave32 op
__device__ __forceinline__ v8f wmma_f32(v2f a, v2f b, v8f c) {
    return __builtin_amdgcn_wmma_f32_16x16x4_f32(
        /*neg_a=*/false, a, /*neg_b=*/false, b,
        /*c_mod=*/(short)0, c, /*reuse_a=*/false, /*reuse_b=*/false);
}

// ---------------------------------------------------------------------------
// Kernel 0: zero the loss accumulators (num, den)
// ---------------------------------------------------------------------------
__global__ void k_init(float* accum) {
    accum[0] = 0.f;
    accum[1] = 0.f;
}

// ---------------------------------------------------------------------------
// Kernel 1: X[b,n,:] = tanh(concat(emb[id[b,n]], box[b,n]) @ W_obj + b_obj)
// X stored as [B*NP_ rows][768], rows with n>=100 zero-filled.
// 28 blocks x 256 threads; each wave owns 16 rows, A in registers (129 x v2f),
// W_obj 16-col panel staged in LDS (float2-interleaved B-layout).
// ---------------------------------------------------------------------------
__global__ __launch_bounds__(256, 1) void k_obj(
    const int*   __restrict__ ids,
    const float* __restrict__ boxes,
    const float* __restrict__ emb,
    const float* __restrict__ Wobj,
    const float* __restrict__ bobj,
    float*       __restrict__ X) {
    __shared__ v2f sB[258 * 16];   // 516 K-rows as 258 float2-pairs x 16 cols (33 KB)

    const int tid  = threadIdx.x;
    const int lane = tid & 31;
    const int w    = tid >> 5;
    const int g    = lane >> 4;    // half-wave group
    const int ln   = lane & 15;
    const int r0   = blockIdx.x * 128 + w * 16;   // first global row of this wave

    // Gather A tile (16 rows x 516) directly into WMMA A-layout registers:
    // lane (g, ln) holds row ln, k = 4j + 2g + {0,1}
    v2f areg[129];
    {
        const int rg = r0 + ln;
        const int ob = rg / NP_;
        const int on = rg % NP_;
        if (on < N_) {
            const float* erow = emb + (size_t)ids[ob * N_ + on] * EMB_;
#pragma unroll
            for (int j = 0; j < 128; ++j)
                areg[j] = *(const v2f*)(erow + 4 * j + 2 * g);
            // k = 512..515 are the 4 box coords (512 % 4 == 0, lands exactly here)
            areg[128] = *(const v2f*)(boxes + (size_t)(ob * N_ + on) * 4 + 2 * g);
        } else {
#pragma unroll
            for (int j = 0; j < 129; ++j) { areg[j].x = 0.f; areg[j].y = 0.f; }
        }
    }

    for (int dc = 0; dc < 48; ++dc) {          // 768 output cols in chunks of 16
        const int d0 = dc * 16;
        __syncthreads();
        for (int idx = tid; idx < 258 * 16; idx += 256) {
            const int j = idx >> 4, n = idx & 15;
            v2f p;
            p.x = Wobj[(size_t)(2 * j)     * HID_ + d0 + n];
            p.y = Wobj[(size_t)(2 * j + 1) * HID_ + d0 + n];
            sB[idx] = p;
        }
        __syncthreads();

        const float bias = bobj[d0 + ln];
        v8f y;
#pragma unroll
        for (int r = 0; r < 8; ++r) y[r] = bias;
#pragma unroll
        for (int ks = 0; ks < 129; ++ks)
            y = wmma_f32(areg[ks], sB[(2 * ks + g) * 16 + ln], y);

        // C-layout store: vgpr r holds row (r + 8g), col = ln
#pragma unroll
        for (int r = 0; r < 8; ++r) {
            const int rr = r0 + r + 8 * g;
            const int nn = rr % NP_;
            X[(size_t)rr * HID_ + d0 + ln] = (nn < N_) ? tanhf(y[r]) : 0.f;
        }
    }
}

// ---------------------------------------------------------------------------
// Kernel 2 (fused): per block = (b, 128 t-rows), per wave = 16 t-rows.
//  for each 16-wide d-chunk (two 384-row K-halves, double-buffered LDS panel):
//    y16 = tanh(dh_tile @ W_pool[:,d0:d0+16] + b_pool)   (192 chained WMMAs)
//    scores[16,112] += y16 @ X[b,:,d0:d0+16]^T           (28 WMMAs)
//  then masked BCE reduction (atomicAdd) and per-row argmax -> pred.
// dh tile lives in registers in A-layout; y never touches HBM. Panel staging
// uses gfx1250 async global->LDS (ASYNCcnt) overlapped with the WMMA chain.
// ---------------------------------------------------------------------------
__global__ __launch_bounds__(256, 1) void k_main(
    const float* __restrict__ dh,
    const float* __restrict__ dmask,
    const int*   __restrict__ targ,
    const float* __restrict__ tmask,
    const float* __restrict__ Wpool,
    const float* __restrict__ bpool,
    const float* __restrict__ X,
    float*       __restrict__ accum,
    float*       __restrict__ out) {
    __shared__ float sB[2][384 * 16];  // double-buffered half-K panel (2 x 24 KB)
    __shared__ float sY[8][16 * 18];   // per-wave y16 transpose buffer, stride 18 (9 KB)

    const int tid  = threadIdx.x;
    const int lane = tid & 31;
    const int w    = tid >> 5;
    const int g    = lane >> 4;
    const int ln   = lane & 15;
    const int b    = blockIdx.x >> 3;
    const int tc   = blockIdx.x & 7;
    const int t0   = tc * 128 + w * 16;      // first t-row of this wave

    // Stage one 384x16 half-panel of W_pool (rows khalf*384.., cols d0..d0+15)
    // into sB[buf] as plain row-major [k][16]; 6 x b128 packets per thread.
    auto stage = [&](int buf, int d0, int khalf) {
        const float* src = Wpool + (size_t)(khalf * 384) * HID_ + d0;
#pragma unroll
        for (int i = 0; i < 6; ++i) {
            const int p   = tid + i * 256;
            const int r   = p >> 2;
            const int seg = (p & 3) * 4;
#ifdef USE_ASYNC_LDS
            __builtin_amdgcn_global_load_async_to_lds_b128(
                (__attribute__((address_space(1))) v4i*)(void*)(src + (size_t)r * HID_ + seg),
                (__attribute__((address_space(3))) v4i*)(void*)(&sB[buf][r * 16 + seg]),
                0, 0);
#else
            const float4 v = *(const float4*)(src + (size_t)r * HID_ + seg);
            *(float4*)(&sB[buf][r * 16 + seg]) = v;
#endif
        }
    };

    // dh tile (16 x 768) -> registers in A-layout: lane (g,ln) row ln, k=4j+2g+{0,1}
    v2f areg[192];
    {
        const float* dhrow = dh + ((size_t)(b * T_ + t0 + ln)) * HID_;
#pragma unroll
        for (int j = 0; j < 192; ++j)
            areg[j] = *(const v2f*)(dhrow + 4 * j + 2 * g);
    }

    v8f acc[7];
#pragma unroll
    for (int nt = 0; nt < 7; ++nt)
#pragma unroll
        for (int r = 0; r < 8; ++r) acc[nt][r] = 0.f;

    float* myY = &sY[w][0];

    // Prologue: kick off the first half-panel (dc=0, khalf=0) into buffer 0.
    stage(0, 0, 0);
    int parity = 0;

    for (int dc = 0; dc < 48; ++dc) {
        const int d0 = dc * 16;

        const float bias = bpool[d0 + ln];
        v8f y;
#pragma unroll
        for (int r = 0; r < 8; ++r) y[r] = bias;

#pragma unroll
        for (int h = 0; h < 2; ++h) {
            // Current buffer's async data arrived (per-wave), then block-wide sync:
            // also guarantees everyone finished reading the *other* buffer, so we
            // can immediately refill it below while computing from this one.
#ifdef USE_ASYNC_LDS
            __builtin_amdgcn_s_wait_asynccnt(0);
#endif
            __syncthreads();

            if (h == 0)           stage(parity ^ 1, d0,      1);   // second half, same chunk
            else if (dc < 47)     stage(parity ^ 1, d0 + 16, 0);   // first half, next chunk

            const float* bufp = &sB[parity][0];
#pragma unroll
            for (int ks = 0; ks < 96; ++ks) {
                v2f bb;
                bb.x = bufp[(4 * ks + 2 * g)     * 16 + ln];
                bb.y = bufp[(4 * ks + 2 * g + 1) * 16 + ln];
                y = wmma_f32(areg[h * 96 + ks], bb, y);
            }
            parity ^= 1;
        }

        // ---- y16 = tanh(y16) ----
#pragma unroll
        for (int r = 0; r < 8; ++r) y[r] = tanhf(y[r]);

        // ---- transpose C-layout -> A-layout via per-wave LDS (same-wave DS is in order) ----
#pragma unroll
        for (int r = 0; r < 8; ++r)
            myY[(r + 8 * g) * 18 + ln] = y[r];

        // ---- scores += y16 @ X[b, :, d0:d0+16]^T   (K = 16 -> 4 k-steps) ----
#pragma unroll
        for (int ks2 = 0; ks2 < 4; ++ks2) {
            const v2f a2 = *(const v2f*)(myY + ln * 18 + 4 * ks2 + 2 * g);
#pragma unroll
            for (int nt = 0; nt < 7; ++nt) {
                const v2f xb = *(const v2f*)(X + ((size_t)(b * NP_ + nt * 16 + ln)) * HID_
                                               + d0 + 4 * ks2 + 2 * g);
                acc[nt] = wmma_f32(a2, xb, acc[nt]);
            }
        }
    }

    // ---- masked BCE: max(s,0) - s*t + log1p(exp(-|s|)) ----
    float dmv[8];
#pragma unroll
    for (int r = 0; r < 8; ++r)
        dmv[r] = dmask[(size_t)b * T_ + t0 + r + 8 * g];

    float num = 0.f, den = 0.f;
#pragma unroll
    for (int nt = 0; nt < 7; ++nt) {
        const int col = nt * 16 + ln;
        const bool cv = (col < N_);
        const float tm = cv ? tmask[b * N_ + col] : 0.f;
        const float tv = cv ? (float)targ[b * N_ + col] : 0.f;
#pragma unroll
        for (int r = 0; r < 8; ++r) {
            const float m = tm * dmv[r];
            const float s = acc[nt][r];
            const float l = fmaxf(s, 0.f) - s * tv + log1pf(expf(-fabsf(s)));
            num += l * m;
            den += m;
        }
    }
    for (int off = 16; off >= 1; off >>= 1) {
        num += __shfl_xor(num, off, 32);
        den += __shfl_xor(den, off, 32);
    }
    if (lane == 0) {
        atomicAdd(&accum[0], num);
        atomicAdd(&accum[1], den);
    }

    // ---- argmax over n (width-16 shuffle: groups hold disjoint row sets) ----
#pragma unroll
    for (int r = 0; r < 8; ++r) {
        float bv = -3.402823466e38f;
        int   bi = 0;
#pragma unroll
        for (int nt = 0; nt < 7; ++nt) {
            const int col = nt * 16 + ln;
            const float s = (col < N_) ? acc[nt][r] : -3.402823466e38f;
            if (s > bv) { bv = s; bi = col; }
        }
#pragma unroll
        for (int msk = 8; msk >= 1; msk >>= 1) {
            const float ov = __shfl_xor(bv, msk, 16);
            const int   oi = __shfl_xor(bi, msk, 16);
            if (ov > bv || (ov == bv && oi < bi)) { bv = ov; bi = oi; }
        }
        if (ln == 0)
            out[1 + (size_t)b * T_ + t0 + r + 8 * g] = (float)bi;
    }
}

// ---------------------------------------------------------------------------
// Kernel 3: finalize loss
// ---------------------------------------------------------------------------
__global__ void k_fin(const float* accum, float* out) {
    out[0] = accum[0] / (accum[1] + 1e-10f);
}

// ---------------------------------------------------------------------------
extern "C" void kernel_launch(void* const* d_in, const int* in_sizes, int n_in,
                              void* d_out, int out_size, void* d_ws, size_t ws_size,
                              hipStream_t stream) {
    const float* dh    = (const float*)d_in[0];
    const float* dmask = (const float*)d_in[1];
    const int*   ids   = (const int*)  d_in[2];
    const float* boxes = (const float*)d_in[3];
    const int*   targ  = (const int*)  d_in[4];
    const float* tmask = (const float*)d_in[5];
    const float* emb   = (const float*)d_in[6];
    const float* Wobj  = (const float*)d_in[7];
    const float* bobj  = (const float*)d_in[8];
    const float* Wpool = (const float*)d_in[9];
    const float* bpool = (const float*)d_in[10];

    float* out   = (float*)d_out;
    float* X     = (float*)d_ws;                          // [B*112][768] fp32 (~10.5 MB)
    float* accum = X + (size_t)B_ * NP_ * HID_;           // 2 floats

    hipLaunchKernelGGL(k_init, dim3(1),   dim3(1),   0, stream, accum);
    hipLaunchKernelGGL(k_obj,  dim3(28),  dim3(256), 0, stream, ids, boxes, emb, Wobj, bobj, X);
    hipLaunchKernelGGL(k_main, dim3(256), dim3(256), 0, stream,
                       dh, dmask, targ, tmask, Wpool, bpool, X, accum, out);
    hipLaunchKernelGGL(k_fin,  dim3(1),   dim3(1),   0, stream, accum, out);
}